// SimpleGIN_90735479095994
// MI455X (gfx1250) — compile-verified
//
#include <hip/hip_runtime.h>

typedef float v2f __attribute__((ext_vector_type(2)));
typedef float v8f __attribute__((ext_vector_type(8)));

#define HID 128
#define PER_BLK 8  // 8 waves of 32 lanes per 256-thread block

// --- embedding gather: x[node,:] = table[idx[node],:] (float4 per lane) ---
__global__ __launch_bounds__(256) void gin_embed(const int* __restrict__ idx,
    const float* __restrict__ table, float* __restrict__ X, int n) {
  int node = blockIdx.x * PER_BLK + (threadIdx.x >> 5);
  int lane = threadIdx.x & 31;
  if (node >= n) return;
  int t = idx[node];
  ((float4*)(X + (size_t)node * HID))[lane] =
      ((const float4*)(table + (size_t)t * HID))[lane];
}

// --- zero fill (grid-stride, float4) ---
__global__ __launch_bounds__(256) void gin_fill0(float4* __restrict__ p, int n4) {
  int i = blockIdx.x * blockDim.x + threadIdx.x;
  int stride = gridDim.x * blockDim.x;
  float4 z = {0.f, 0.f, 0.f, 0.f};
  for (; i < n4; i += stride) p[i] = z;
}

// --- edge-list SpMM scatter: agg[dst,:] += x[src,:], one wave per edge ---
__global__ __launch_bounds__(256) void gin_scatter(const int* __restrict__ src,
    const int* __restrict__ dst, const float* __restrict__ X,
    float* Agg, int n_edges) {
  int e = blockIdx.x * PER_BLK + (threadIdx.x >> 5);
  int lane = threadIdx.x & 31;
  if (e >= n_edges) return;
  int s = src[e], d = dst[e];
  float4 v = ((const float4*)(X + (size_t)s * HID))[lane];
  float* p = Agg + (size_t)d * HID + lane * 4;
  atomicAdd(p + 0, v.x);
  atomicAdd(p + 1, v.y);
  atomicAdd(p + 2, v.z);
  atomicAdd(p + 3, v.w);
}

// --- agg += (1+eps) * x (elementwise, float4, grid-stride) ---
__global__ __launch_bounds__(256) void gin_combine(float4* __restrict__ Agg,
    const float4* __restrict__ X, const float* __restrict__ eps, int n4) {
  int i = blockIdx.x * blockDim.x + threadIdx.x;
  int stride = gridDim.x * blockDim.x;
  float c = 1.0f + eps[0];
  for (; i < n4; i += stride) {
    float4 a = Agg[i], x = X[i];
    a.x += c * x.x; a.y += c * x.y; a.z += c * x.z; a.w += c * x.w;
    Agg[i] = a;
  }
}

// --- Y[n x 128] = X[n x 128] @ W[128 x 128] + bias (optional ReLU) ---
// One wave computes a 16-row strip across all eight 16-col tiles with
// V_WMMA_F32_16X16X4_F32.
// W is staged in LDS row-pair interleaved: lw[p*256 + col*2 + (k&1)] holds
// W[k=2p+{0,1}, col], so each B fragment (W[k,col], W[k+1,col]) is one
// 8B-aligned contiguous ds_load_b64 directly into the WMMA operand pair.
// A loads are software-pipelined one k-step ahead to hide VMEM latency.
__global__ __launch_bounds__(256) void gin_gemm128(const float* __restrict__ X,
    const float* __restrict__ W, const float* __restrict__ bias,
    float* __restrict__ Y, int n_rows, int relu) {
  __shared__ float lw[HID * HID];  // 64 KB of the 320 KB/WGP LDS
  for (int i = threadIdx.x; i < (HID / 2) * HID; i += 256) {
    int p = i >> 7;    // row pair index 0..63
    int c = i & 127;   // column
    float2 v;
    v.x = W[(size_t)(2 * p) * HID + c];
    v.y = W[(size_t)(2 * p + 1) * HID + c];
    *(float2*)(lw + p * 256 + c * 2) = v;
  }
  __syncthreads();

  const int lane = threadIdx.x & 31;
  const int wave = threadIdx.x >> 5;
  const int row0 = (blockIdx.x * 8 + wave) * 16;
  if (row0 >= n_rows) return;

  const int lhalf = lane & 15;
  const int khi   = (lane >> 4) << 1;  // lanes 0-15 -> K+0/1, lanes 16-31 -> K+2/3

  v8f acc[8];
#pragma unroll
  for (int n = 0; n < 8; ++n)
    acc[n] = (v8f){0.f, 0.f, 0.f, 0.f, 0.f, 0.f, 0.f, 0.f};

  const float* arow = X + (size_t)(row0 + lhalf) * HID + khi;
  v2f a = *(const v2f*)arow;  // k-step 0 fragment
#pragma unroll
  for (int ks = 0; ks < HID / 4; ++ks) {
    v2f an = a;
    if (ks + 1 < HID / 4) an = *(const v2f*)(arow + (ks + 1) * 4);  // prefetch
    const float* bp = lw + ((ks * 4 + khi) >> 1) * 256 + lhalf * 2;
#pragma unroll
    for (int n = 0; n < 8; ++n) {
      v2f b = *(const v2f*)(bp + n * 32);  // ds_load_b64, pair-contiguous
      acc[n] = __builtin_amdgcn_wmma_f32_16x16x4_f32(
          false, a, false, b, (short)0, acc[n], false, false);
    }
    a = an;
  }

  const int mbase = row0 + ((lane >> 4) << 3);  // +8 rows for upper half-wave
#pragma unroll
  for (int n = 0; n < 8; ++n) {
    int col = n * 16 + lhalf;
    float bv = bias[col];
#pragma unroll
    for (int r = 0; r < 8; ++r) {
      float v = acc[n][r] + bv;
      if (relu) v = fmaxf(v, 0.f);
      Y[(size_t)(mbase + r) * HID + col] = v;
    }
  }
}

// --- fused final GEMV (128 -> 1) + per-graph pooling ---
__global__ __launch_bounds__(256) void gin_out(const float* __restrict__ H,
    const float* __restrict__ w2, const float* __restrict__ b2,
    const int* __restrict__ batch, float* out, int n) {
  int node = blockIdx.x * PER_BLK + (threadIdx.x >> 5);
  int lane = threadIdx.x & 31;
  if (node >= n) return;
  float4 h = ((const float4*)(H + (size_t)node * HID))[lane];
  float4 w = ((const float4*)w2)[lane];
  float s = h.x * w.x + h.y * w.y + h.z * w.z + h.w * w.w;
#pragma unroll
  for (int off = 16; off > 0; off >>= 1) s += __shfl_down(s, off, 32);
  if (lane == 0) {
    float v = s + b2[0];
    out[2048 + node] = v;               // node scores (x [N,1])
    atomicAdd(&out[batch[node]], v);    // y_hat segment_sum
  }
}

extern "C" void kernel_launch(void* const* d_in, const int* in_sizes, int n_in,
                              void* d_out, int out_size, void* d_ws, size_t ws_size,
                              hipStream_t stream) {
  (void)n_in; (void)out_size; (void)ws_size;
  const int*   x_idx = (const int*)d_in[0];
  const int*   e_src = (const int*)d_in[1];
  const int*   e_dst = (const int*)d_in[2];
  const int*   batch = (const int*)d_in[3];
  const float* embed = (const float*)d_in[4];
  const float* eps0  = (const float*)d_in[5];
  const float* w1_0  = (const float*)d_in[6];
  const float* b1_0  = (const float*)d_in[7];
  const float* w2_0  = (const float*)d_in[8];
  const float* b2_0  = (const float*)d_in[9];
  const float* eps1  = (const float*)d_in[10];
  const float* w1_1  = (const float*)d_in[11];
  const float* b1_1  = (const float*)d_in[12];
  const float* w2_1  = (const float*)d_in[13];
  const float* b2_1  = (const float*)d_in[14];
  float* out = (float*)d_out;

  const int N = in_sizes[0];
  const int E = in_sizes[1];
  const size_t NM = (size_t)N * HID;

  float* xbuf = (float*)d_ws;     // node features        (51.2 MB)
  float* agg  = xbuf + NM;        // aggregation buffer   (51.2 MB)
  float* hbuf = agg + NM;         // GEMM intermediate    (51.2 MB)

  dim3 blk(256);
  int node_blocks = (N + PER_BLK - 1) / PER_BLK;
  int edge_blocks = (E + PER_BLK - 1) / PER_BLK;
  int gemm_blocks = (N / 16 + 7) / 8;
  int n4 = (int)(NM / 4);
  int ew_blocks = 2048;

  gin_embed<<<node_blocks, blk, 0, stream>>>(x_idx, embed, xbuf, N);

  // ---- layer 0: agg = A@x ; h = agg + (1+eps0)x ; x = relu((h W1 + b1) W2 + b2)
  gin_fill0<<<ew_blocks, blk, 0, stream>>>((float4*)agg, n4);
  gin_scatter<<<edge_blocks, blk, 0, stream>>>(e_src, e_dst, xbuf, agg, E);
  gin_combine<<<ew_blocks, blk, 0, stream>>>((float4*)agg, (const float4*)xbuf, eps0, n4);
  gin_gemm128<<<gemm_blocks, blk, 0, stream>>>(agg, w1_0, b1_0, hbuf, N, 0);
  gin_gemm128<<<gemm_blocks, blk, 0, stream>>>(hbuf, w2_0, b2_0, xbuf, N, 1);

  // ---- layer 1: agg = A@x ; h = agg + (1+eps1)x ; h1 = h W1 + b1
  gin_fill0<<<ew_blocks, blk, 0, stream>>>((float4*)agg, n4);
  gin_scatter<<<edge_blocks, blk, 0, stream>>>(e_src, e_dst, xbuf, agg, E);
  gin_combine<<<ew_blocks, blk, 0, stream>>>((float4*)agg, (const float4*)xbuf, eps1, n4);
  gin_gemm128<<<gemm_blocks, blk, 0, stream>>>(agg, w1_1, b1_1, hbuf, N, 0);

  // ---- readout: node score + graph pooling
  gin_fill0<<<8, blk, 0, stream>>>((float4*)out, 2048 / 4);
  gin_out<<<node_blocks, blk, 0, stream>>>(hbuf, w2_1, b2_1, batch, out, N);
}